// EnasController_75265006895188
// MI455X (gfx1250) — compile-verified
//
#include <hip/hip_runtime.h>
#include <hip/hip_bf16.h>
#include <math.h>

#define HID 64
#define NL 12
#define NB 6
#define TANH_C 1.5f
#define SKIP_T 0.8f

typedef __attribute__((ext_vector_type(2))) float v2f;
typedef __attribute__((ext_vector_type(8))) float v8f;

// Explicit global-address-space pointers: keeps weight traffic on
// global_load_b64 (saddr + voffset) instead of flat_load with 64-bit VALU
// address math.
typedef const __attribute__((address_space(1))) float *gfp;
typedef const __attribute__((address_space(1))) v2f *gv2p;

// ---------------- threefry2x32 (JAX PRNG) ----------------
__device__ __forceinline__ unsigned rotl(unsigned x, unsigned n) {
  return (x << n) | (x >> (32u - n));
}

__device__ void tf2x32(unsigned k0, unsigned k1, unsigned c0, unsigned c1,
                       unsigned &o0, unsigned &o1) {
  unsigned ks0 = k0, ks1 = k1, ks2 = k0 ^ k1 ^ 0x1BD11BDAu;
  unsigned x0 = c0 + ks0, x1 = c1 + ks1;
#define R4(a, b, c, d)                                                         \
  x0 += x1; x1 = rotl(x1, a); x1 ^= x0;                                        \
  x0 += x1; x1 = rotl(x1, b); x1 ^= x0;                                        \
  x0 += x1; x1 = rotl(x1, c); x1 ^= x0;                                        \
  x0 += x1; x1 = rotl(x1, d); x1 ^= x0;
  R4(13, 15, 26, 6)  x0 += ks1; x1 += ks2 + 1u;
  R4(17, 29, 16, 24) x0 += ks2; x1 += ks0 + 2u;
  R4(13, 15, 26, 6)  x0 += ks0; x1 += ks1 + 3u;
  R4(17, 29, 16, 24) x0 += ks1; x1 += ks2 + 4u;
  R4(13, 15, 26, 6)  x0 += ks2; x1 += ks0 + 5u;
#undef R4
  o0 = x0; o1 = x1;
}

__device__ __forceinline__ float bits_to_u01(unsigned bits) {
  union { unsigned u; float f; } cv;
  cv.u = (bits >> 9) | 0x3f800000u;
  float f = cv.f - 1.0f;                 // [0,1)
  const float tiny = 1.1754943508e-38f;  // FLT_MIN (JAX minval)
  f = f * (1.0f - tiny) + tiny;
  return fmaxf(tiny, f);
}

__device__ __forceinline__ float gmb(unsigned bits) {
  return -logf(-logf(bits_to_u01(bits)));
}

__device__ __forceinline__ float sigm(float x) { return 1.0f / (1.0f + expf(-x)); }

// ---------------- gate matvec via WMMA f32 16x16x4 ----------------
// g[0..255] = W_ih @ x + W_hh @ h   (biases added later)
// 8 waves, wave w owns row-tiles 2w and 2w+1 (16 rows each).
// A fragment: lane l holds W[tile*16 + l%16, 4j + 2*(l/16) + {0,1}]
// B fragment: x broadcast over all 16 N-columns -> every D column = matvec.
__device__ __forceinline__ void compute_gates(gfp w_ih, gfp w_hh,
                                              const float *xv /*LDS*/,
                                              const float *hv /*LDS*/,
                                              float *g_lds /*LDS*/) {
  const int tid = threadIdx.x;
  const int wave = tid >> 5;
  const int lane = tid & 31;
  const int rlo = lane & 15;
  const int hi = lane >> 4;   // 0: lanes 0-15, 1: lanes 16-31
  const int khalf = hi << 1;  // K offset 0 or 2 (A/B VGPR layout)

#pragma unroll
  for (int tt = 0; tt < 2; ++tt) {
    const int tile = wave * 2 + tt;
    const int row = tile * 16 + rlo;
    v8f acc = {};
    gfp wri = w_ih + row * HID + khalf;
    gfp wrh = w_hh + row * HID + khalf;
#pragma unroll
    for (int j = 0; j < 16; ++j) {  // K = 0..63 : W_ih @ x
      v2f a = *(gv2p)(wri + j * 4);
      v2f b = *(const v2f *)(xv + j * 4 + khalf);
      acc = __builtin_amdgcn_wmma_f32_16x16x4_f32(false, a, false, b,
                                                  (short)0, acc, false, false);
    }
#pragma unroll
    for (int j = 0; j < 16; ++j) {  // K = 64..127 : W_hh @ h
      v2f a = *(gv2p)(wrh + j * 4);
      v2f b = *(const v2f *)(hv + j * 4 + khalf);
      acc = __builtin_amdgcn_wmma_f32_16x16x4_f32(false, a, false, b,
                                                  (short)0, acc, false, false);
    }
    // Extract column 0 of D (all columns identical):
    // lanes 0-15 vgpr v -> M=v ; lanes 16-31 vgpr v -> M=v+8.
    if (rlo == 0) {
      const int base = tile * 16 + (hi << 3);
#pragma unroll
      for (int v = 0; v < 8; ++v) g_lds[base + v] = acc[v];
    }
  }
}

// threads 0..63: LSTM gate nonlinearity + state update (gate order i,f,g,o)
__device__ __forceinline__ void cell_update(const float *g /*LDS*/, gfp b_ih,
                                            gfp b_hh, float *h, float *c) {
  const int j = threadIdx.x;
  if (j < HID) {
    float gi = g[j]           + b_ih[j]           + b_hh[j];
    float gf = g[HID + j]     + b_ih[HID + j]     + b_hh[HID + j];
    float gg = g[2 * HID + j] + b_ih[2 * HID + j] + b_hh[2 * HID + j];
    float go = g[3 * HID + j] + b_ih[3 * HID + j] + b_hh[3 * HID + j];
    float cn = sigm(gf) * c[j] + sigm(gi) * tanhf(gg);
    c[j] = cn;
    h[j] = sigm(go) * tanhf(cn);
  }
}

__global__ __launch_bounds__(256) void enas_controller_kernel(
    const float *__restrict__ g_emb_, const float *__restrict__ w_emb_,
    const float *__restrict__ w_soft_, const float *__restrict__ w_attn2_,
    const float *__restrict__ v_attn_, const float *__restrict__ w_ih0_,
    const float *__restrict__ w_hh0_, const float *__restrict__ b_ih0_,
    const float *__restrict__ b_hh0_, const float *__restrict__ w_ih1_,
    const float *__restrict__ w_hh1_, const float *__restrict__ b_ih1_,
    const float *__restrict__ b_hh1_, const int *__restrict__ seed_p_,
    float *__restrict__ out_) {
  gfp g_emb = (gfp)g_emb_, w_emb = (gfp)w_emb_, w_soft = (gfp)w_soft_;
  gfp w_attn2 = (gfp)w_attn2_, v_attn = (gfp)v_attn_;
  gfp w_ih0 = (gfp)w_ih0_, w_hh0 = (gfp)w_hh0_;
  gfp b_ih0 = (gfp)b_ih0_, b_hh0 = (gfp)b_hh0_;
  gfp w_ih1 = (gfp)w_ih1_, w_hh1 = (gfp)w_hh1_;
  gfp b_ih1 = (gfp)b_ih1_, b_hh1 = (gfp)b_hh1_;

  __shared__ __align__(16) float s_x[HID];
  __shared__ __align__(16) float s_h0[HID], s_c0[HID], s_h1[HID], s_c1[HID];
  __shared__ __align__(16) float s_g[4 * HID];
  __shared__ __align__(16) float s_anchor[NL][HID];
  __shared__ __align__(16) float s_aw[NL][HID];
  __shared__ __align__(16) float s_vec[HID];
  __shared__ float s_q[NL];
  __shared__ float s_logit[NB];
  __shared__ float s_scale;
  __shared__ int s_outid;
  __shared__ int s_skip[NL];

  const int tid = threadIdx.x;

  // Warm L2: prefetch the four weight matrices (global_prefetch_b8).
  {
    const float *mats[4] = {w_ih0_, w_hh0_, w_ih1_, w_hh1_};
#pragma unroll
    for (int m = 0; m < 4; ++m)
      for (int ofs = tid * 32; ofs < 4 * HID * HID; ofs += 256 * 32)
        __builtin_prefetch(mats[m] + ofs, 0, 0);
  }

  if (tid < HID) {
    s_h0[tid] = 0.0f; s_c0[tid] = 0.0f; s_h1[tid] = 0.0f; s_c1[tid] = 0.0f;
    s_x[tid] = g_emb[tid];
  }
  __syncthreads();

  // thread-0 persistent state
  unsigned key0 = 0, key1 = 0;
  unsigned kb0 = 0, kb1 = 0, ks0 = 0, ks1 = 0;
  float lp_sum = 0.0f, ent_sum = 0.0f, kl_sum = 0.0f;
  if (tid == 0) {
    unsigned long long s = (unsigned long long)(long long)seed_p_[0];
    key0 = (unsigned)(s >> 32);
    key1 = (unsigned)(s & 0xFFFFFFFFull);
  }

  // Stacked 2-cell LSTM step; "out" ends up in s_h1.
  auto lstm_step = [&](const float *xin) {
#pragma unroll 1
    for (int cell = 0; cell < 2; ++cell) {
      gfp wih = cell ? w_ih1 : w_ih0;
      gfp whh = cell ? w_hh1 : w_hh0;
      gfp bih = cell ? b_ih1 : b_ih0;
      gfp bhh = cell ? b_hh1 : b_hh0;
      const float *xv = cell ? (const float *)s_h0 : xin;
      float *hv = cell ? s_h1 : s_h0;
      float *cv = cell ? s_c1 : s_c0;
      compute_gates(wih, whh, xv, hv, s_g);
      __syncthreads();
      cell_update(s_g, bih, bhh, hv, cv);
      __syncthreads();
    }
  };

  for (int lid = 0; lid < NL; ++lid) {
    // key, kb, ks = split(key, 3): counts [0..5] -> pairs (0,3),(1,4),(2,5)
    if (tid == 0) {
      unsigned a0, b0, a1, b1, a2, b2;
      tf2x32(key0, key1, 0u, 3u, a0, b0);
      tf2x32(key0, key1, 1u, 4u, a1, b1);
      tf2x32(key0, key1, 2u, 5u, a2, b2);
      key0 = a0; key1 = a1;  // bits = [a0,a1,a2,b0,b1,b2] -> (3,2)
      kb0 = a2;  kb1 = b0;
      ks0 = b1;  ks1 = b2;
    }

    lstm_step(s_x);

    // logits = w_soft @ out
    if (tid < NB) {
      float acc = 0.0f;
      for (int i = 0; i < HID; ++i) acc += w_soft[tid * HID + i] * s_h1[i];
      s_logit[tid] = acc;
    }
    __syncthreads();

    // branch sampling (categorical = Gumbel-argmax), log-prob, entropy
    if (tid == 0) {
      unsigned A0, B0, A1, B1, A2, B2;
      tf2x32(kb0, kb1, 0u, 3u, A0, B0);
      tf2x32(kb0, kb1, 1u, 4u, A1, B1);
      tf2x32(kb0, kb1, 2u, 5u, A2, B2);
      unsigned flat[NB] = {A0, A1, A2, B0, B1, B2};
      float m = -INFINITY;
      for (int i = 0; i < NB; ++i) m = fmaxf(m, s_logit[i]);
      float se = 0.0f;
      for (int i = 0; i < NB; ++i) se += expf(s_logit[i] - m);
      float lse = m + logf(se);
      float best = -INFINITY; int bi = 0; float ent = 0.0f;
      for (int i = 0; i < NB; ++i) {
        float lp = s_logit[i] - lse;
        ent -= expf(lp) * lp;
        float z = s_logit[i] + gmb(flat[i]);
        if (z > best) { best = z; bi = i; }
      }
      lp_sum += s_logit[bi] - lse;
      ent_sum += ent;
      s_outid = bi;
    }
    __syncthreads();

    // inputs = w_emb[out_id]
    if (tid < HID) s_x[tid] = w_emb[s_outid * HID + tid];
    __syncthreads();

    lstm_step(s_x);

    if (lid > 0) {
      // aw_cur = out @ w_attn2^T  (row j of w_attn2 dot out)
      if (tid < HID) {
        float acc = 0.0f;
        for (int i = 0; i < HID; ++i) acc += w_attn2[tid * HID + i] * s_h1[i];
        s_vec[tid] = acc;
      }
      __syncthreads();

      // q_l = tanh(anchors_w1[l] + aw_cur) . v_attn
      if (tid < lid) {
        float q = 0.0f;
        for (int i = 0; i < HID; ++i)
          q += tanhf(s_aw[tid][i] + s_vec[i]) * v_attn[i];
        s_q[tid] = q;
      }
      __syncthreads();

      // skip sampling, log-probs, entropy, KL vs SKIP_TARGET
      if (tid == 0) {
        unsigned abuf[NL], bbuf[NL];
        for (int i = 0; i < lid; ++i)
          tf2x32(ks0, ks1, (unsigned)i, (unsigned)(lid + i), abuf[i], bbuf[i]);
        auto flat = [&](int j) { return j < lid ? abuf[j] : bbuf[j - lid]; };
        int sc = 0;
        for (int l = 0; l < lid; ++l) {
          float q = s_q[l];
          float L0 = TANH_C * tanhf(q);
          float L1 = TANH_C * tanhf(-q);
          float g0 = gmb(flat(2 * l));
          float g1 = gmb(flat(2 * l + 1));
          int sk = (L1 + g1 > L0 + g0) ? 1 : 0;  // argmax, first-on-tie
          float m = fmaxf(L0, L1);
          float lse = m + logf(expf(L0 - m) + expf(L1 - m));
          float lp0 = L0 - lse, lp1 = L1 - lse;
          lp_sum += sk ? lp1 : lp0;
          ent_sum -= expf(lp0) * lp0 + expf(lp1) * lp1;
          float sp0 = sigm(L0), sp1 = sigm(L1);
          kl_sum += sp0 * logf(sp0 / SKIP_T) + sp1 * logf(sp1 / SKIP_T);
          sc += sk;
          s_skip[l] = sk;
        }
        s_scale = 1.0f + (float)sc;
      }
      __syncthreads();

      // inputs = (skips @ anchors) / (1 + sc); append anchor & anchor_w1
      if (tid < HID) {
        float acc = 0.0f;
        for (int l = 0; l < lid; ++l)
          if (s_skip[l]) acc += s_anchor[l][tid];
        s_x[tid] = acc / s_scale;
        s_anchor[lid][tid] = s_h1[tid];
        s_aw[lid][tid] = s_vec[tid];
      }
      __syncthreads();
    } else {
      // lid == 0: inputs = g_emb[0]; anchors_w1[0] computed directly
      if (tid < HID) {
        s_x[tid] = g_emb[tid];
        s_anchor[0][tid] = s_h1[tid];
        float acc = 0.0f;
        for (int i = 0; i < HID; ++i) acc += w_attn2[tid * HID + i] * s_h1[i];
        s_aw[0][tid] = acc;
      }
      __syncthreads();
    }
  }

  if (tid == 0) {
    out_[0] = lp_sum;
    out_[1] = ent_sum;
    out_[2] = kl_sum;
  }
}

extern "C" void kernel_launch(void *const *d_in, const int *in_sizes, int n_in,
                              void *d_out, int out_size, void *d_ws,
                              size_t ws_size, hipStream_t stream) {
  (void)in_sizes; (void)n_in; (void)out_size; (void)d_ws; (void)ws_size;
  const float *g_emb   = (const float *)d_in[0];
  const float *w_emb   = (const float *)d_in[1];
  const float *w_soft  = (const float *)d_in[2];
  const float *w_attn2 = (const float *)d_in[3];
  const float *v_attn  = (const float *)d_in[4];
  const float *w_ih0   = (const float *)d_in[5];
  const float *w_hh0   = (const float *)d_in[6];
  const float *b_ih0   = (const float *)d_in[7];
  const float *b_hh0   = (const float *)d_in[8];
  const float *w_ih1   = (const float *)d_in[9];
  const float *w_hh1   = (const float *)d_in[10];
  const float *b_ih1   = (const float *)d_in[11];
  const float *b_hh1   = (const float *)d_in[12];
  const int   *seed    = (const int *)d_in[13];

  enas_controller_kernel<<<1, 256, 0, stream>>>(
      g_emb, w_emb, w_soft, w_attn2, v_attn, w_ih0, w_hh0, b_ih0, b_hh0,
      w_ih1, w_hh1, b_ih1, b_hh1, seed, (float *)d_out);
}